// CanCountLeaveOperator_37993280700433
// MI455X (gfx1250) — compile-verified
//
#include <hip/hip_runtime.h>
#include <hip/hip_bf16.h>

typedef __attribute__((ext_vector_type(2))) float v2f;
typedef __attribute__((ext_vector_type(8))) float v8f;

#define MAXN 4096
#define TILE 16

// Async global->LDS copy of 16 bytes (CDNA5, tracked by ASYNCcnt).
__device__ __forceinline__ void async_b128(unsigned int lds_off, unsigned long long gaddr) {
    asm volatile("global_load_async_to_lds_b128 %0, %1, off"
                 :: "v"(lds_off), "v"(gaddr) : "memory");
}

// Tile epilogue: prime test + 24 stores. DIAG=false -> no per-lane guard at all.
template <bool DIAG>
__device__ __forceinline__ void emit_tile(
    const int* __restrict__ s_p, const float* __restrict__ s_rp,
    float* __restrict__ outL, float* __restrict__ outC, float* __restrict__ outP,
    const v8f& dL, const v8f& dC,
    int i_start, int j, int n)
{
    const int   b  = s_p[j];
    const float fb = (float)b;
    const float rb = s_rp[j];

    int i = i_start;
    long long base = (long long)i * n - (((long long)i * (i - 1)) >> 1);

    #pragma unroll
    for (int r = 0; r < 8; ++r) {
        const long long o = base + (j - i);
        const int   a  = s_p[i];
        const float fa = (float)a;
        // exact for 1 <= a,b < 1000: divisible <=> a == b*round(a*rcp(b))
        const int qab = (int)__builtin_rintf(fa * rb);
        const int qba = (int)__builtin_rintf(fb * s_rp[i]);
        const int dv  = (int)((a - qab * b) == 0) | (int)((b - qba * a) == 0);
        const float pv = dv ? (float)(a > b ? a : b) : 0.0f;
        if (!DIAG || j >= i) {
            outL[o] = dL[r];
            outC[o] = dC[r];
            outP[o] = pv;
        }
        base += (long long)(n - i);   // base(i+1) = base(i) + (n - i)
        ++i;
    }
}

__global__ __launch_bounds__(256) void triu_pairs_kernel(
    const int* __restrict__ xl, const float* __restrict__ xc,
    const int* __restrict__ xp, float* __restrict__ out,
    int n, long long N, int TB)
{
    const int bi = blockIdx.y;
    // Whole block below the diagonal: exit before paying for staging.
    if ((int)blockIdx.x * 8 + 7 < bi) return;

    // [0,16K): leaves  [16K,32K): cplx  [32K,48K): prime  [48K,64K): rcp(prime)
    __shared__ __align__(16) unsigned char smem[MAXN * 16];
    int*   s_l  = (int*)smem;
    float* s_c  = (float*)(smem + MAXN * 4);
    int*   s_p  = (int*)(smem + MAXN * 8);
    float* s_rp = (float*)(smem + MAXN * 12);

    const int tid = threadIdx.x;

    // ---- Stage inputs into LDS with async copies (ASYNCcnt path) ----
    unsigned int lds0 = (unsigned int)(unsigned long long)(void*)smem;
    unsigned long long gl = (unsigned long long)xl;
    unsigned long long gc = (unsigned long long)xc;
    unsigned long long gp = (unsigned long long)xp;
    const int nq = n >> 2;                      // 16-byte packets per array
    for (int k = tid; k < nq; k += blockDim.x) {
        unsigned int off = (unsigned int)(k << 4);
        async_b128(lds0 + off,            gl + off);
        async_b128(lds0 + MAXN * 4 + off, gc + off);
        async_b128(lds0 + MAXN * 8 + off, gp + off);
    }
    asm volatile("s_wait_asynccnt 0x0" ::: "memory");
    __syncthreads();

    // Reciprocal table for exact small-int divisibility test.
    for (int k = tid; k < n; k += blockDim.x)
        s_rp[k] = __builtin_amdgcn_rcpf((float)s_p[k]);
    __syncthreads();

    // ---- One 16x16 tile per wave ----
    const int bj = blockIdx.x * 8 + (tid >> 5);
    if (bj < bi || bj >= TB) return;            // wave-uniform -> EXEC stays full

    const int lane = tid & 31;
    const int half = lane >> 4;                 // 0: lanes 0-15, 1: lanes 16-31
    const int l15  = lane & 15;
    const int i0 = bi * TILE, j0 = bj * TILE;

    // A = [x_i | 1 | 0 | 0] (16x4), B = [1 ; x_j ; 0 ; 0] (4x16), C = 1
    // => D[M][N] = x[i0+M] + x[j0+N] + 1
    v2f aC, bC, aL, bL;
    if (half == 0) {
        aC[0] = s_c[i0 + l15];        aC[1] = 1.0f;          // A: K=0,1 columns
        bC[0] = 1.0f;                 bC[1] = s_c[j0 + l15]; // B: K=0,1 rows
        aL[0] = (float)s_l[i0 + l15]; aL[1] = 1.0f;
        bL[0] = 1.0f;                 bL[1] = (float)s_l[j0 + l15];
    } else {                                                 // K=2,3 -> zeros
        aC[0] = aC[1] = bC[0] = bC[1] = 0.0f;
        aL[0] = aL[1] = bL[0] = bL[1] = 0.0f;
    }
    v8f c1 = {1.0f, 1.0f, 1.0f, 1.0f, 1.0f, 1.0f, 1.0f, 1.0f};

    v8f dC = __builtin_amdgcn_wmma_f32_16x16x4_f32(false, aC, false, bC,
                                                   (short)0, c1, false, false);
    v8f dL = __builtin_amdgcn_wmma_f32_16x16x4_f32(false, aL, false, bL,
                                                   (short)0, c1, false, false);

    float* __restrict__ outL = out;
    float* __restrict__ outC = out + N;
    float* __restrict__ outP = out + 2 * N;

    const int i_start = i0 + (half << 3);       // D layout: vgpr r -> M = r + 8*half
    const int j = j0 + l15;

    if (bi == bj) {
        emit_tile<true >(s_p, s_rp, outL, outC, outP, dL, dC, i_start, j, n);
    } else {
        emit_tile<false>(s_p, s_rp, outL, outC, outP, dL, dC, i_start, j, n);
    }
}

extern "C" void kernel_launch(void* const* d_in, const int* in_sizes, int n_in,
                              void* d_out, int out_size, void* d_ws, size_t ws_size,
                              hipStream_t stream) {
    const int*   xl = (const int*)d_in[0];
    const float* xc = (const float*)d_in[1];
    const int*   xp = (const int*)d_in[2];
    float* out = (float*)d_out;

    const int n = in_sizes[0];                       // 4096
    const long long N = (long long)n * (n + 1) / 2;  // 8,390,656
    const int TB = n / TILE;                         // 256 tile bands

    dim3 grid((TB + 7) / 8, TB, 1);                  // 8 waves (tiles) per block
    triu_pairs_kernel<<<grid, dim3(256), 0, stream>>>(xl, xc, xp, out, n, N, TB);
}